// CalibrationLoss_34041910788289
// MI455X (gfx1250) — compile-verified
//
#include <hip/hip_runtime.h>

// CalibrationLoss (ECE-style) for MI455X / gfx1250.
//
// Roofline: B*C*4 = 400MB mandatory reads -> ~17us floor at 23.3 TB/s.
// Bandwidth-bound streaming + 15-bin histogram. Primary read engine: gfx1250
// async global->LDS copies (ASYNCcnt) staging 32-row tiles per wave with
// double buffering, consumed via ds_load_b128 (400B LDS row stride is
// bank-conflict-free for b128 on 64 banks). Fallback: direct float4 streaming.

#define NUM_BINS 15

#if defined(__has_builtin)
#  if __has_builtin(__builtin_amdgcn_global_load_async_to_lds_b128)
#    define HAVE_ASYNC_LDS 1
#  else
#    define HAVE_ASYNC_LDS 0
#  endif
#else
#  define HAVE_ASYNC_LDS 0
#endif

// Builtin expects pointer-to-int4-vector in AS1 (global) / AS3 (LDS).
typedef int v4i __attribute__((vector_size(16)));
typedef __attribute__((address_space(1))) v4i* g4p;
typedef __attribute__((address_space(3))) v4i* l4p;

#if defined(__has_builtin) && __has_builtin(__builtin_amdgcn_s_wait_asynccnt)
#  define WAIT_ASYNCCNT(n)                                   \
    do {                                                     \
        asm volatile("" ::: "memory");                       \
        __builtin_amdgcn_s_wait_asynccnt(n);                 \
        asm volatile("" ::: "memory");                       \
    } while (0)
#else
#  define WAIT_ASYNCCNT(n)                                   \
    asm volatile("s_wait_asynccnt %0" ::"i"(n) : "memory")
#endif

// ---------------------------------------------------------------------------
__global__ void calib_init(double* __restrict__ g_conf,
                           unsigned int* __restrict__ g_cnt,
                           unsigned int* __restrict__ g_acc) {
    int i = threadIdx.x;
    if (i < NUM_BINS) {
        g_conf[i] = 0.0;
        g_cnt[i]  = 0u;
        g_acc[i]  = 0u;
    }
}

// ---------------------------------------------------------------------------
// Per-row bin update into block-local LDS histogram.
__device__ __forceinline__ void bin_update(float maxv, int maxi, int target,
                                           double* s_conf, unsigned int* s_cnt,
                                           unsigned int* s_acc) {
    int bin = (int)ceilf(maxv * (float)NUM_BINS) - 1;  // same f32 ops as reference
    bin = bin < 0 ? 0 : (bin > NUM_BINS - 1 ? NUM_BINS - 1 : bin);
    atomicAdd(&s_cnt[bin], 1u);
    if (maxi == target) atomicAdd(&s_acc[bin], 1u);
    unsafeAtomicAdd(&s_conf[bin], (double)maxv);
}

__device__ __forceinline__ void merge_hist(const double* s_conf,
                                           const unsigned int* s_cnt,
                                           const unsigned int* s_acc,
                                           double* g_conf, unsigned int* g_cnt,
                                           unsigned int* g_acc) {
    if (threadIdx.x < NUM_BINS) {
        unsigned int c = s_cnt[threadIdx.x];
        if (c) atomicAdd(&g_cnt[threadIdx.x], c);
        unsigned int a = s_acc[threadIdx.x];
        if (a) atomicAdd(&g_acc[threadIdx.x], a);
        double cf = s_conf[threadIdx.x];
        if (cf != 0.0) unsafeAtomicAdd(&g_conf[threadIdx.x], cf);
    }
}

#if HAVE_ASYNC_LDS
// ---------------------------------------------------------------------------
// Async staging kernel (C == 100 only). 4 waves/block; each wave owns a
// 2 x 12.8KB double-buffered staging region. Tile = 32 consecutive rows.
#define AW            4                      // waves per block
#define TILE_BYTES    12800                  // 32 rows * 400B
#define WAVE_STAGE    (2 * TILE_BYTES)       // double buffer

__global__ __launch_bounds__(AW * 32) void calib_hist_async100(
    const float* __restrict__ probs,
    const int* __restrict__ targets,
    double* __restrict__ g_conf,
    unsigned int* __restrict__ g_cnt,
    unsigned int* __restrict__ g_acc,
    int B)
{
    __shared__ alignas(16) char stage[AW * WAVE_STAGE];
    __shared__ double           s_conf[NUM_BINS];
    __shared__ unsigned int     s_cnt[NUM_BINS];
    __shared__ unsigned int     s_acc[NUM_BINS];
    if (threadIdx.x < NUM_BINS) {
        s_conf[threadIdx.x] = 0.0;
        s_cnt[threadIdx.x]  = 0u;
        s_acc[threadIdx.x]  = 0u;
    }
    __syncthreads();

    const int lane     = threadIdx.x & 31;
    const int waveId   = threadIdx.x >> 5;
    const int gWave    = blockIdx.x * AW + waveId;
    const int nWaves   = gridDim.x * AW;
    const int numTiles = B / 32;

    char* myStage = stage + waveId * WAVE_STAGE;
    const unsigned long long gBase =
        (unsigned long long)probs + (unsigned long long)lane * 16ull;
    // Low 32 bits of a generic LDS pointer == LDS byte address (ISA flat rules).
    const unsigned int ldsBase =
        (unsigned int)(unsigned long long)(myStage) + (unsigned int)(lane * 16);

    // Issue all 25 async b128 copies for one tile into buffer `buf`.
    // Offsets are baked into the pointers (builtin imm offset kept 0).
    auto issue = [&](int tile, int buf) {
        unsigned long long g = gBase + (unsigned long long)tile * TILE_BYTES;
        unsigned int       l = ldsBase + (unsigned int)(buf * TILE_BYTES);
        #pragma unroll
        for (int i = 0; i < 25; ++i) {
            __builtin_amdgcn_global_load_async_to_lds_b128(
                (g4p)(g + (unsigned long long)(i * 512)),
                (l4p)(l + (unsigned int)(i * 512)), 0, 0);
        }
    };

    int buf = 0;
    if (gWave < numTiles) issue(gWave, 0);

    for (int tile = gWave; tile < numTiles; tile += nWaves) {
        int next = tile + nWaves;
        if (next < numTiles) {
            issue(next, buf ^ 1);
            WAIT_ASYNCCNT(25);   // retire the older tile's 25 in-order loads
        } else {
            WAIT_ASYNCCNT(0);
        }

        const float4* __restrict__ lrow =
            (const float4*)(myStage + buf * TILE_BYTES + lane * 400);
        float maxv = -1.0f;
        int   maxi = 0;
        #pragma unroll 5
        for (int i = 0; i < 25; ++i) {
            float4 v = lrow[i];
            if (v.x > maxv) { maxv = v.x; maxi = 4 * i + 0; }
            if (v.y > maxv) { maxv = v.y; maxi = 4 * i + 1; }
            if (v.z > maxv) { maxv = v.z; maxi = 4 * i + 2; }
            if (v.w > maxv) { maxv = v.w; maxi = 4 * i + 3; }
        }
        int row = tile * 32 + lane;
        bin_update(maxv, maxi, targets[row], s_conf, s_cnt, s_acc);
        buf ^= 1;
    }

    // Tail rows (B % 32 != 0): direct loads, handled by block 0.
    {
        int row = numTiles * 32 + blockIdx.x * blockDim.x + threadIdx.x;
        if (blockIdx.x == 0 && row < B) {
            const float4* __restrict__ rp =
                (const float4*)(probs + (size_t)row * 100);
            float maxv = -1.0f;
            int   maxi = 0;
            for (int i = 0; i < 25; ++i) {
                float4 v = rp[i];
                if (v.x > maxv) { maxv = v.x; maxi = 4 * i + 0; }
                if (v.y > maxv) { maxv = v.y; maxi = 4 * i + 1; }
                if (v.z > maxv) { maxv = v.z; maxi = 4 * i + 2; }
                if (v.w > maxv) { maxv = v.w; maxi = 4 * i + 3; }
            }
            bin_update(maxv, maxi, targets[row], s_conf, s_cnt, s_acc);
        }
    }
    __syncthreads();
    merge_hist(s_conf, s_cnt, s_acc, g_conf, g_cnt, g_acc);
}
#endif  // HAVE_ASYNC_LDS

// ---------------------------------------------------------------------------
// Direct-load kernel (fallback when async builtins unavailable).
template <int C>
__global__ __launch_bounds__(256) void calib_hist_fixedC(
    const float* __restrict__ probs,
    const int* __restrict__ targets,
    double* __restrict__ g_conf,
    unsigned int* __restrict__ g_cnt,
    unsigned int* __restrict__ g_acc,
    int B)
{
    __shared__ double       s_conf[NUM_BINS];
    __shared__ unsigned int s_cnt[NUM_BINS];
    __shared__ unsigned int s_acc[NUM_BINS];
    if (threadIdx.x < NUM_BINS) {
        s_conf[threadIdx.x] = 0.0;
        s_cnt[threadIdx.x]  = 0u;
        s_acc[threadIdx.x]  = 0u;
    }
    __syncthreads();

    const int stride = gridDim.x * blockDim.x;
    for (int row = blockIdx.x * blockDim.x + threadIdx.x; row < B; row += stride) {
        const float4* __restrict__ rp =
            reinterpret_cast<const float4*>(probs + (size_t)row * C);
        int nextRow = row + stride;
        if (nextRow < B) {
            __builtin_prefetch(probs + (size_t)nextRow * C, 0, 3);
        }
        float maxv = -1.0f;
        int   maxi = 0;
        #pragma unroll 5
        for (int i = 0; i < C / 4; ++i) {
            float4 v = rp[i];
            if (v.x > maxv) { maxv = v.x; maxi = 4 * i + 0; }
            if (v.y > maxv) { maxv = v.y; maxi = 4 * i + 1; }
            if (v.z > maxv) { maxv = v.z; maxi = 4 * i + 2; }
            if (v.w > maxv) { maxv = v.w; maxi = 4 * i + 3; }
        }
        bin_update(maxv, maxi, targets[row], s_conf, s_cnt, s_acc);
    }
    __syncthreads();
    merge_hist(s_conf, s_cnt, s_acc, g_conf, g_cnt, g_acc);
}

// Generic-C fallback (any C / any alignment).
__global__ __launch_bounds__(256) void calib_hist_generic(
    const float* __restrict__ probs,
    const int* __restrict__ targets,
    double* __restrict__ g_conf,
    unsigned int* __restrict__ g_cnt,
    unsigned int* __restrict__ g_acc,
    int B, int C)
{
    __shared__ double       s_conf[NUM_BINS];
    __shared__ unsigned int s_cnt[NUM_BINS];
    __shared__ unsigned int s_acc[NUM_BINS];
    if (threadIdx.x < NUM_BINS) {
        s_conf[threadIdx.x] = 0.0;
        s_cnt[threadIdx.x]  = 0u;
        s_acc[threadIdx.x]  = 0u;
    }
    __syncthreads();

    const int stride = gridDim.x * blockDim.x;
    for (int row = blockIdx.x * blockDim.x + threadIdx.x; row < B; row += stride) {
        const float* __restrict__ rp = probs + (size_t)row * C;
        float maxv = -1.0f;
        int   maxi = 0;
        for (int i = 0; i < C; ++i) {
            float v = rp[i];
            if (v > maxv) { maxv = v; maxi = i; }
        }
        bin_update(maxv, maxi, targets[row], s_conf, s_cnt, s_acc);
    }
    __syncthreads();
    merge_hist(s_conf, s_cnt, s_acc, g_conf, g_cnt, g_acc);
}

// ---------------------------------------------------------------------------
__global__ void calib_final(const double* __restrict__ g_conf,
                            const unsigned int* __restrict__ g_cnt,
                            const unsigned int* __restrict__ g_acc,
                            float* __restrict__ out, int B)
{
    if (threadIdx.x == 0 && blockIdx.x == 0) {
        double n = (double)B;
        double mmce = 0.0;
        for (int i = 0; i < NUM_BINS; ++i) {
            double c = (double)g_cnt[i];
            if (c > 0.0) {
                double acc_mean  = (double)g_acc[i] / c;
                double conf_mean = g_conf[i] / c;
                mmce += (c / n) * fabs(acc_mean - conf_mean);
            }
        }
        out[0] = (float)mmce;
    }
}

// ---------------------------------------------------------------------------
extern "C" void kernel_launch(void* const* d_in, const int* in_sizes, int n_in,
                              void* d_out, int out_size, void* d_ws, size_t ws_size,
                              hipStream_t stream) {
    const float* probs   = (const float*)d_in[0];
    const int*   targets = (const int*)d_in[1];
    float*       out     = (float*)d_out;

    const int B = in_sizes[1];
    const int C = (B > 0) ? (in_sizes[0] / B) : 0;

    double*       g_conf = (double*)d_ws;
    unsigned int* g_cnt  = (unsigned int*)(g_conf + NUM_BINS);
    unsigned int* g_acc  = g_cnt + NUM_BINS;

    calib_init<<<1, 64, 0, stream>>>(g_conf, g_cnt, g_acc);

    if (C == 100) {
#if HAVE_ASYNC_LDS
        calib_hist_async100<<<2048, AW * 32, 0, stream>>>(
            probs, targets, g_conf, g_cnt, g_acc, B);
#else
        calib_hist_fixedC<100><<<2048, 256, 0, stream>>>(
            probs, targets, g_conf, g_cnt, g_acc, B);
#endif
    } else {
        calib_hist_generic<<<2048, 256, 0, stream>>>(
            probs, targets, g_conf, g_cnt, g_acc, B, C);
    }

    calib_final<<<1, 32, 0, stream>>>(g_conf, g_cnt, g_acc, out, B);
}